// FusionAttentionConv3D_31464930410847
// MI455X (gfx1250) — compile-verified
//
#include <hip/hip_runtime.h>
#include <hip/hip_bf16.h>

// ---------------------------------------------------------------------------
// Problem constants (match reference)
// ---------------------------------------------------------------------------
#define BN   2
#define CDIM 48
#define HH   384
#define WW   384
#define NPIX (HH * WW)          // 147456
#define HEADS 8
#define HC    (CDIM / HEADS)    // 6

typedef __attribute__((ext_vector_type(2))) float v2f;
typedef __attribute__((ext_vector_type(8))) float v8f;

// ---------------------------------------------------------------------------
// Pointwise (1x1) conv as GEMM: out[m][pix] = sum_k W[m][k] * in[k][pix]
// V_WMMA_F32_16X16X4_F32, MT M-tiles of 16 rows kept per wave so the B
// (activation) fragment is loaded ONCE per K-step and reused MT times.
//   A (16x4 f32):  lane<16: vgpr j = A[M=lane][K=k0+j];  lane>=16: K=k0+2+j
//   B (4x16 f32):  lane l, vgpr j -> (K = k0 + j + 2*(l>>4), N = l&15)
//   C (16x16 f32): vgpr j, lanes 0-15 -> M=j; lanes 16-31 -> M=j+8; N=lane&15
// Grid: (Npix/128, B); block = 256 threads = 8 waves, 16 pixels per wave.
// ---------------------------------------------------------------------------
template <int MT>
__global__ __launch_bounds__(256)
void pw_wmma(const float* __restrict__ in, const float* __restrict__ W,
             float* __restrict__ out, int Cin,
             size_t inStride, size_t wStride, size_t outStride)
{
    const int lane = threadIdx.x & 31;
    const int wave = threadIdx.x >> 5;
    const int lo   = lane & 15;
    const int hi   = lane >> 4;
    const int pix  = blockIdx.x * 128 + wave * 16 + lo;
    const int b    = blockIdx.y;

    const float* inb  = in  + (size_t)b * inStride;
    const float* Wb   = W   + (size_t)b * wStride;
    float*       outb = out + (size_t)b * outStride;

    const v8f zero = {0.f, 0.f, 0.f, 0.f, 0.f, 0.f, 0.f, 0.f};
    v8f acc[MT];
    #pragma unroll
    for (int mt = 0; mt < MT; ++mt) acc[mt] = zero;

    for (int k0 = 0; k0 < Cin; k0 += 4) {
        const int kk = k0 + 2 * hi;
        // prefetch next K-step's activation lines into cache
        if (k0 + 4 < Cin) {
            __builtin_prefetch(&inb[(size_t)(kk + 4) * NPIX + pix], 0, 3);
            __builtin_prefetch(&inb[(size_t)(kk + 5) * NPIX + pix], 0, 3);
        }
        v2f bb;
        bb.x = inb[(size_t)kk * NPIX + pix];
        bb.y = inb[(size_t)(kk + 1) * NPIX + pix];
        #pragma unroll
        for (int mt = 0; mt < MT; ++mt) {
            const int m = mt * 16 + lo;
            v2f a;
            a.x = Wb[m * Cin + kk];
            a.y = Wb[m * Cin + kk + 1];
            acc[mt] = __builtin_amdgcn_wmma_f32_16x16x4_f32(
                false, a, false, bb, (short)0, acc[mt], false, false);
        }
    }

    #pragma unroll
    for (int mt = 0; mt < MT; ++mt)
        #pragma unroll
        for (int j = 0; j < 8; ++j) {
            const int mm = mt * 16 + j + 8 * hi;
            outb[(size_t)mm * NPIX + pix] = acc[mt][j];
        }
}

// ---------------------------------------------------------------------------
// wave32 all-lane sum reduction
// ---------------------------------------------------------------------------
__device__ __forceinline__ float waveRedSum(float v)
{
    #pragma unroll
    for (int off = 16; off >= 1; off >>= 1)
        v += __shfl_xor(v, off, 32);
    return v;
}

// 3x3 stencil at (y,x) with zero padding; w9 = 9 weights in LDS
__device__ __forceinline__ float tap9(const float* __restrict__ base,
                                      const float* __restrict__ w9,
                                      int y, int x)
{
    float acc = 0.f;
    #pragma unroll
    for (int ky = 0; ky < 3; ++ky) {
        const int yy = y + ky - 1;
        if (yy < 0 || yy >= HH) continue;
        const float* rp = base + (size_t)yy * WW;
        #pragma unroll
        for (int kx = 0; kx < 3; ++kx) {
            const int xx = x + kx - 1;
            if (xx < 0 || xx >= WW) continue;
            acc = fmaf(w9[ky * 3 + kx], rp[xx], acc);
        }
    }
    return acc;
}

// ---------------------------------------------------------------------------
// Depthwise 3x3 (center depth slice of the 3x3x3 kernels) on q1 (48ch) and
// kv1 (96ch). Produces:
//   qT, kT : pixel-major [b][pix][48]   (lane-contiguous frags for qk^T WMMA)
//   v      : channel-major [b][48][pix] (B-matrix layout for final GEMM)
//   sumsq  : per (b, channel) sum of squares, q in 0..47, k in 48..95
// Grid: (Npix/256, B), block 256.
// ---------------------------------------------------------------------------
__global__ __launch_bounds__(256)
void dw3x3(const float* __restrict__ q1, const float* __restrict__ kv1,
           const float* __restrict__ Wq2, const float* __restrict__ Wkv2,
           float* __restrict__ qT, float* __restrict__ kT,
           float* __restrict__ vv, float* __restrict__ sumsq)
{
    const int tid = threadIdx.x;
    const int b   = blockIdx.y;
    const int pix = blockIdx.x * 256 + tid;
    const int y   = pix / WW;
    const int x   = pix % WW;

    __shared__ float wq[48 * 9];
    __shared__ float wkv[96 * 9];
    __shared__ float ssq[96];

    // kd = 1 slice of the (C,1,3,3,3) depthwise weights
    for (int i = tid; i < 48 * 9; i += 256) wq[i]  = Wq2[(i / 9) * 27 + 9 + (i % 9)];
    for (int i = tid; i < 96 * 9; i += 256) wkv[i] = Wkv2[(i / 9) * 27 + 9 + (i % 9)];
    if (tid < 96) ssq[tid] = 0.f;
    __syncthreads();

    const float* qb = q1  + (size_t)b * 48 * NPIX;
    const float* kb = kv1 + (size_t)b * 96 * NPIX;

    float row[48];

    // ---- q path -----------------------------------------------------------
    #pragma unroll
    for (int c = 0; c < 48; ++c) {
        const float acc = tap9(qb + (size_t)c * NPIX, &wq[c * 9], y, x);
        row[c] = acc;
        const float r = waveRedSum(acc * acc);
        if ((tid & 31) == 0) atomicAdd(&ssq[c], r);
    }
    {
        float4* dst = (float4*)(qT + ((size_t)b * NPIX + pix) * 48);
        #pragma unroll
        for (int i = 0; i < 12; ++i)
            dst[i] = make_float4(row[4*i], row[4*i+1], row[4*i+2], row[4*i+3]);
    }

    // ---- k path (kv channels 0..47) --------------------------------------
    #pragma unroll
    for (int c = 0; c < 48; ++c) {
        const float acc = tap9(kb + (size_t)c * NPIX, &wkv[c * 9], y, x);
        row[c] = acc;
        const float r = waveRedSum(acc * acc);
        if ((tid & 31) == 0) atomicAdd(&ssq[48 + c], r);
    }
    {
        float4* dst = (float4*)(kT + ((size_t)b * NPIX + pix) * 48);
        #pragma unroll
        for (int i = 0; i < 12; ++i)
            dst[i] = make_float4(row[4*i], row[4*i+1], row[4*i+2], row[4*i+3]);
    }

    // ---- v path (kv channels 48..95), channel-major ----------------------
    #pragma unroll
    for (int c = 0; c < 48; ++c) {
        const float acc = tap9(kb + (size_t)(48 + c) * NPIX, &wkv[(48 + c) * 9], y, x);
        vv[((size_t)b * 48 + c) * NPIX + pix] = acc;
    }

    __syncthreads();
    if (tid < 96) atomicAdd(&sumsq[b * 96 + tid], ssq[tid]);
}

// ---------------------------------------------------------------------------
// S[b][c][d] = sum_pix q[c][pix] * k[d][pix]   (48x48 GEMM, K = Npix)
// qT/kT are pixel-major so WMMA fragment loads are lane-contiguous.
// Each wave owns 2048 pixels of K with 9 WMMA accumulators (3x3 tiles);
// partials combined via global f32 atomics. Grid: (Npix/16384, B).
// ---------------------------------------------------------------------------
#define QK_PIX_PER_WAVE 2048
__global__ __launch_bounds__(256)
void qk_wmma(const float* __restrict__ qT, const float* __restrict__ kT,
             float* __restrict__ S)
{
    const int lane = threadIdx.x & 31;
    const int wave = threadIdx.x >> 5;
    const int lo   = lane & 15;
    const int hi   = lane >> 4;
    const int b    = blockIdx.y;
    const size_t base = (size_t)b * NPIX;

    const int p0 = blockIdx.x * (8 * QK_PIX_PER_WAVE) + wave * QK_PIX_PER_WAVE;

    const v8f zero = {0.f,0.f,0.f,0.f,0.f,0.f,0.f,0.f};
    v8f acc[3][3];
    #pragma unroll
    for (int i = 0; i < 3; ++i)
        #pragma unroll
        for (int j = 0; j < 3; ++j) acc[i][j] = zero;

    for (int kp = p0; kp < p0 + QK_PIX_PER_WAVE; kp += 4) {
        const float* qrow0 = qT + (base + kp + 2 * hi) * 48;
        const float* qrow1 = qrow0 + 48;
        const float* krow0 = kT + (base + kp + 2 * hi) * 48;
        const float* krow1 = krow0 + 48;
        // prefetch next K-step rows (stream at HBM rate)
        __builtin_prefetch(qrow0 + 4 * 48, 0, 3);
        __builtin_prefetch(krow0 + 4 * 48, 0, 3);

        v2f a[3], bb[3];
        #pragma unroll
        for (int t = 0; t < 3; ++t) {
            a[t].x  = qrow0[t * 16 + lo];
            a[t].y  = qrow1[t * 16 + lo];
            bb[t].x = krow0[t * 16 + lo];
            bb[t].y = krow1[t * 16 + lo];
        }
        #pragma unroll
        for (int mt = 0; mt < 3; ++mt)
            #pragma unroll
            for (int nt = 0; nt < 3; ++nt)
                acc[mt][nt] = __builtin_amdgcn_wmma_f32_16x16x4_f32(
                    false, a[mt], false, bb[nt], (short)0, acc[mt][nt], false, false);
    }

    #pragma unroll
    for (int mt = 0; mt < 3; ++mt)
        #pragma unroll
        for (int nt = 0; nt < 3; ++nt)
            #pragma unroll
            for (int j = 0; j < 8; ++j) {
                const int r   = mt * 16 + j + 8 * hi;
                const int col = nt * 16 + lo;
                atomicAdd(&S[b * 2304 + r * 48 + col], acc[mt][nt][j]);
            }
}

// ---------------------------------------------------------------------------
// Per batch: norms, softmax over head-local 6x6 blocks, then fold
// M = Wout * BlockDiag(attn)  (48x48).  Grid: (B), block 256.
// ---------------------------------------------------------------------------
__global__ __launch_bounds__(256)
void attn_mat(const float* __restrict__ S, const float* __restrict__ sumsq,
              const float* __restrict__ temp, const float* __restrict__ Wout,
              float* __restrict__ Mf)
{
    const int b = blockIdx.x;
    const int tid = threadIdx.x;
    __shared__ float A48[48 * 48];
    __shared__ float nq[48], nk[48];

    for (int i = tid; i < 2304; i += 256) A48[i] = 0.f;
    if (tid < 48)      nq[tid]      = fmaxf(sqrtf(sumsq[b * 96 + tid]), 1e-12f);
    else if (tid < 96) nk[tid - 48] = fmaxf(sqrtf(sumsq[b * 96 + tid]), 1e-12f);
    __syncthreads();

    if (tid < 48) {
        const int h = tid / HC;
        const int base = h * HC;
        const float t = temp[h];
        float l[HC];
        float mx = -1e30f;
        #pragma unroll
        for (int d = 0; d < HC; ++d) {
            l[d] = S[b * 2304 + tid * 48 + base + d] / (nq[tid] * nk[base + d]) * t;
            mx = fmaxf(mx, l[d]);
        }
        float se = 0.f;
        #pragma unroll
        for (int d = 0; d < HC; ++d) { l[d] = __expf(l[d] - mx); se += l[d]; }
        const float inv = 1.f / se;
        #pragma unroll
        for (int d = 0; d < HC; ++d) A48[tid * 48 + base + d] = l[d] * inv;
    }
    __syncthreads();

    for (int i = tid; i < 2304; i += 256) {
        const int o = i / 48;
        const int d = i % 48;
        float acc = 0.f;
        #pragma unroll
        for (int c2 = 0; c2 < 48; ++c2)
            acc = fmaf(Wout[o * 48 + c2], A48[c2 * 48 + d], acc);
        Mf[b * 2304 + i] = acc;
    }
}

// ---------------------------------------------------------------------------
// Launch orchestration
// inputs: 0=img 1=evs 2=Wq1 3=Wq2 4=Wkv1 5=Wkv2 6=Wout 7=temperature
// ---------------------------------------------------------------------------
extern "C" void kernel_launch(void* const* d_in, const int* in_sizes, int n_in,
                              void* d_out, int out_size, void* d_ws, size_t ws_size,
                              hipStream_t stream)
{
    const float* img  = (const float*)d_in[0];
    const float* evs  = (const float*)d_in[1];
    const float* Wq1  = (const float*)d_in[2];
    const float* Wq2  = (const float*)d_in[3];
    const float* Wkv1 = (const float*)d_in[4];
    const float* Wkv2 = (const float*)d_in[5];
    const float* Wout = (const float*)d_in[6];
    const float* temp = (const float*)d_in[7];
    float* out = (float*)d_out;

    const size_t chN = (size_t)CDIM * NPIX;   // 48 * 147456

    float* ws  = (float*)d_ws;
    float* q1  = ws;                          // B*48*N
    float* kv1 = q1  + (size_t)BN * chN;      // B*96*N
    float* qT  = kv1 + (size_t)BN * 2 * chN;  // B*N*48
    float* kT  = qT  + (size_t)BN * chN;      // B*N*48
    float* vv  = kT  + (size_t)BN * chN;      // B*48*N
    float* S   = vv  + (size_t)BN * chN;      // B*48*48
    float* ssq = S   + (size_t)BN * 2304;     // B*96
    float* Mf  = ssq + (size_t)BN * 96;       // B*48*48

    // zero the accumulators (S and ssq are contiguous)
    (void)hipMemsetAsync(S, 0, (size_t)(BN * 2304 + BN * 96) * sizeof(float), stream);

    dim3 blk(256);

    // q1 = Wq1 * evs      (Cout = 48 -> 3 M-tiles per wave)
    pw_wmma<3><<<dim3(NPIX / 128, BN), blk, 0, stream>>>(
        evs, Wq1, q1, CDIM, chN, 0, chN);

    // kv1 = Wkv1 * img    (Cout = 96 -> 6 M-tiles per wave)
    pw_wmma<6><<<dim3(NPIX / 128, BN), blk, 0, stream>>>(
        img, Wkv1, kv1, CDIM, chN, 0, 2 * chN);

    // depthwise 3x3 -> qT, kT, v, sum-of-squares
    dw3x3<<<dim3(NPIX / 256, BN), blk, 0, stream>>>(
        q1, kv1, Wq2, Wkv2, qT, kT, vv, ssq);

    // S = q k^T (WMMA reduction over all pixels)
    qk_wmma<<<dim3(NPIX / (8 * QK_PIX_PER_WAVE), BN), blk, 0, stream>>>(qT, kT, S);

    // softmax + fold Wout -> per-batch 48x48 matrix Mf
    attn_mat<<<dim3(BN), blk, 0, stream>>>(S, ssq, temp, Wout, Mf);

    // out = Mf * v  (per-batch weights via wStride = 2304, inStride = chN)
    pw_wmma<3><<<dim3(NPIX / 128, BN), blk, 0, stream>>>(
        vv, Mf, out, CDIM, chN, 2304, chN);
}